// UcpeSelfAttention_34256659153140
// MI455X (gfx1250) — compile-verified
//
#include <hip/hip_runtime.h>
#include <math.h>

typedef __attribute__((ext_vector_type(2))) float v2f;
typedef __attribute__((ext_vector_type(8))) float v8f;
typedef __attribute__((ext_vector_type(4))) unsigned int u32x4;
typedef __attribute__((ext_vector_type(8))) int i32x8;
typedef __attribute__((ext_vector_type(4))) int i32x4;

#define DIMC 1536
#define NHEADS 24
#define HDIM 64
#define BATCH 2
#define NCAM 16
#define HWTOK 64
#define TOK 1024               // tokens per batch (NCAM * HWTOK)
#define BT (BATCH * TOK)       // 2048
#define DP 32                  // projective half of head dim

#define KCHUNK 32              // K elements staged per TDM transfer
#define NKCH (DIMC / KCHUNK)   // 48 chunks
#define WROWS 64               // weight rows (N) per block tile
#define WSTRIDE 34             // padded LDS row stride in dwords (32 + 2 pad)

__device__ __forceinline__ v8f zero8() {
  v8f z;
#pragma unroll
  for (int i = 0; i < 8; i++) z[i] = 0.0f;
  return z;
}

__device__ __forceinline__ v8f wmma4(v2f a, v2f b, v8f c) {
  // D = A(16x4, f32) * B(4x16, f32) + C(16x16, f32)
  return __builtin_amdgcn_wmma_f32_16x16x4_f32(false, a, false, b, (short)0, c,
                                               false, false);
}

// Issue one TDM load of a 64-row x 32-dword weight tile into LDS, with the TDM
// inserting 2 pad dwords after every 128-byte row (LDS stride = 34 dwords).
__device__ __forceinline__ void tdm_load_wtile(unsigned lds_addr,
                                               unsigned long long gaddr,
                                               i32x8 g1, i32x4 gz4, i32x8 gz8) {
  u32x4 g0;
  g0[0] = 1u;                                  // count=1, user mode, no gather
  g0[1] = lds_addr;                            // LDS byte address
  g0[2] = (unsigned)gaddr;                     // global_addr[31:0]
  g0[3] = (unsigned)((gaddr >> 32) & 0x01FFFFFFu) | 0x80000000u; // [56:32] | type=2
  // clang-23 toolchain: 6-arg form (g0, g1, g2, g3, g4, cpol)
  __builtin_amdgcn_tensor_load_to_lds(g0, g1, gz4, gz4, gz8, 0);
}

// ---------------------------------------------------------------------------
// Build per-camera P = [[K * E[:3,:4]],[0,0,0,1]] and its closed-form inverse.
// ---------------------------------------------------------------------------
__global__ void prep_mats(const float* __restrict__ vm, const float* __restrict__ Kin,
                          float* __restrict__ P, float* __restrict__ Pinv) {
  int cam = blockIdx.x * blockDim.x + threadIdx.x;
  if (cam >= BATCH * NCAM) return;
  const float* E = vm + cam * 16;
  const float* Km = Kin + cam * 9;
  float P3[3][4];
#pragma unroll
  for (int i = 0; i < 3; i++)
#pragma unroll
    for (int j = 0; j < 4; j++) {
      float s = 0.0f;
#pragma unroll
      for (int k = 0; k < 3; k++) s += Km[i * 3 + k] * E[k * 4 + j];
      P3[i][j] = s;
    }
  float* Pp = P + cam * 16;
#pragma unroll
  for (int i = 0; i < 3; i++)
#pragma unroll
    for (int j = 0; j < 4; j++) Pp[i * 4 + j] = P3[i][j];
  Pp[12] = 0.f; Pp[13] = 0.f; Pp[14] = 0.f; Pp[15] = 1.f;

  float a00 = P3[0][0], a01 = P3[0][1], a02 = P3[0][2];
  float a10 = P3[1][0], a11 = P3[1][1], a12 = P3[1][2];
  float a20 = P3[2][0], a21 = P3[2][1], a22 = P3[2][2];
  float C00 = a11 * a22 - a12 * a21, C01 = a12 * a20 - a10 * a22, C02 = a10 * a21 - a11 * a20;
  float C10 = a02 * a21 - a01 * a22, C11 = a00 * a22 - a02 * a20, C12 = a01 * a20 - a00 * a21;
  float C20 = a01 * a12 - a02 * a11, C21 = a02 * a10 - a00 * a12, C22 = a00 * a11 - a01 * a10;
  float det = a00 * C00 + a01 * C01 + a02 * C02;
  float id = 1.0f / det;
  float inv[3][3];
  inv[0][0] = C00 * id; inv[0][1] = C10 * id; inv[0][2] = C20 * id;
  inv[1][0] = C01 * id; inv[1][1] = C11 * id; inv[1][2] = C21 * id;
  inv[2][0] = C02 * id; inv[2][1] = C12 * id; inv[2][2] = C22 * id;
  float b0 = P3[0][3], b1 = P3[1][3], b2 = P3[2][3];
  float* Pi = Pinv + cam * 16;
#pragma unroll
  for (int i = 0; i < 3; i++) {
    Pi[i * 4 + 0] = inv[i][0];
    Pi[i * 4 + 1] = inv[i][1];
    Pi[i * 4 + 2] = inv[i][2];
    Pi[i * 4 + 3] = -(inv[i][0] * b0 + inv[i][1] * b1 + inv[i][2] * b2);
  }
  Pi[12] = 0.f; Pi[13] = 0.f; Pi[14] = 0.f; Pi[15] = 1.f;
}

// ---------------------------------------------------------------------------
// C[BT, DIMC] = X[BT, DIMC] * W[DIMC, DIMC]^T + bias.  One wave => 16x64 tile.
// Weight K-chunks (64 rows x 32 dwords) are DMA'd into LDS by the Tensor Data
// Mover, double buffered on TENSORcnt; inner loop is ds_load + v_wmma.
// A-frag: a[v] = X[m][4c + 2*hi + v]; B-frag: b[v] = Wlds[n][4c + 2*hi + v].
// ---------------------------------------------------------------------------
__global__ void __launch_bounds__(32) gemm_xwt(const float* __restrict__ X,
                                               const float* __restrict__ W,
                                               const float* __restrict__ bias,
                                               float* __restrict__ C) {
  __shared__ float wlds[2][WROWS * WSTRIDE];
  const int lane = threadIdx.x;
  const int m = lane & 15, hi = lane >> 4;
  const int mbase = blockIdx.x * 16;
  const int nbase = blockIdx.y * 64;

  const float* xr = X + (size_t)(mbase + m) * DIMC + 2 * hi;

  // ---- TDM descriptor group1 (uniform constants) ----
  i32x8 g1;
  g1[0] = (2 << 16)          // data_size = 4 bytes
        | (1 << 20)          // pad_enable
        | (4 << 22)          // pad_interval: 16 x 8B = one 128B row
        | (1 << 25);         // pad_amount: 2 dwords
  g1[1] = (int)(KCHUNK << 16);        // tensor_dim0 = 32
  g1[2] = (int)(WROWS << 16);         // tensor_dim1 = 64
  g1[3] = (int)(KCHUNK << 16);        // tile_dim0 = 32
  g1[4] = (int)WROWS;                 // tile_dim1 = 64, tile_dim2 = 0
  g1[5] = (int)DIMC;                  // tensor_dim0_stride = 1536
  g1[6] = 0;
  g1[7] = 0;
  i32x4 gz4; gz4[0] = 0; gz4[1] = 0; gz4[2] = 0; gz4[3] = 0;
  i32x8 gz8;
#pragma unroll
  for (int i = 0; i < 8; i++) gz8[i] = 0;

  unsigned lds_a[2];
  lds_a[0] = (unsigned)(unsigned long long)&wlds[0][0];
  lds_a[1] = (unsigned)(unsigned long long)&wlds[1][0];
  unsigned long long wbase =
      (unsigned long long)(const void*)(W + (size_t)nbase * DIMC);

  v8f acc[4];
#pragma unroll
  for (int j = 0; j < 4; j++) acc[j] = zero8();

  // prologue: stage chunk 0
  tdm_load_wtile(lds_a[0], wbase, g1, gz4, gz8);

#pragma unroll 1
  for (int ch = 0; ch < NKCH; ch++) {
    if (ch + 1 < NKCH) {
      tdm_load_wtile(lds_a[(ch + 1) & 1],
                     wbase + (unsigned long long)(ch + 1) * (KCHUNK * 4), g1, gz4, gz8);
      __builtin_amdgcn_s_wait_tensorcnt((short)1);  // chunk ch landed
    } else {
      __builtin_amdgcn_s_wait_tensorcnt((short)0);
    }
    asm volatile("" ::: "memory");

    const float* wb = &wlds[ch & 1][0];
    const float* xc = xr + (size_t)ch * KCHUNK;
#pragma unroll
    for (int cc = 0; cc < KCHUNK / 4; cc++) {
      v2f a = *(const v2f*)(xc + 4 * cc);
#pragma unroll
      for (int j = 0; j < 4; j++) {
        v2f b = *(const v2f*)(wb + (j * 16 + m) * WSTRIDE + 4 * cc + 2 * hi);
        acc[j] = wmma4(a, b, acc[j]);
      }
    }
    asm volatile("" ::: "memory");
  }

#pragma unroll
  for (int j = 0; j < 4; j++) {
    int n = nbase + j * 16 + m;
    float bv = bias[n];
#pragma unroll
    for (int r = 0; r < 8; r++) {
      int row = mbase + r + 8 * hi;
      C[(size_t)row * DIMC + n] = acc[j][r] + bv;
    }
  }
}

// ---------------------------------------------------------------------------
// Per-(token, head) UCPE transform: 8x projective 4x4 blocks + 16 rotary pairs.
// transposeM=1 applies M^T (query side), sinSign=-1 inverts rotation (output).
// ---------------------------------------------------------------------------
__global__ void transform_tokens(float* __restrict__ X, const float* __restrict__ mats,
                                 int transposeM, float sinSign) {
  int idx = blockIdx.x * blockDim.x + threadIdx.x;
  if (idx >= BATCH * TOK * NHEADS) return;
  int h = idx % NHEADS;
  int t = (idx / NHEADS) % TOK;
  int b = idx / (NHEADS * TOK);
  int cam = b * NCAM + t / HWTOK;
  int hw = t % HWTOK;

  const float* M = mats + cam * 16;
  float Mm[16];
#pragma unroll
  for (int i = 0; i < 4; i++)
#pragma unroll
    for (int j = 0; j < 4; j++)
      Mm[i * 4 + j] = transposeM ? M[j * 4 + i] : M[i * 4 + j];

  float* p = X + (size_t)(b * TOK + t) * DIMC + h * HDIM;

  float xb[DP];
#pragma unroll
  for (int i = 0; i < DP; i++) xb[i] = p[i];
#pragma unroll
  for (int blk = 0; blk < DP / 4; blk++) {
#pragma unroll
    for (int i = 0; i < 4; i++) {
      float y = Mm[i * 4 + 0] * xb[blk * 4 + 0] + Mm[i * 4 + 1] * xb[blk * 4 + 1] +
                Mm[i * 4 + 2] * xb[blk * 4 + 2] + Mm[i * 4 + 3] * xb[blk * 4 + 3];
      p[blk * 4 + i] = y;
    }
  }

  // patch-center coords; angles th = coord * 100^(-f/8)
  float u = ((float)(hw & 7) + 0.5f) * 16.0f;   // IW/PX = 16
  float v = ((float)(hw >> 3) + 0.5f) * 16.0f;  // IH/PY = 16
  const float LN100_8 = 0.57564627324851142f;   // ln(100)/8
#pragma unroll
  for (int pi = 0; pi < 16; pi++) {
    int f = pi & 7;
    float axis = (pi < 8) ? u : v;
    float th = axis * expf(-(float)f * LN100_8);
    float cth = cosf(th);
    float sth = sinSign * sinf(th);
    float a = p[DP + 2 * pi];
    float bb = p[DP + 2 * pi + 1];
    p[DP + 2 * pi] = cth * a - sth * bb;
    p[DP + 2 * pi + 1] = sth * a + cth * bb;
  }
}

// ---------------------------------------------------------------------------
// Flash attention: one wave per (b, h, 16-query tile). fp32 WMMA throughout.
// ---------------------------------------------------------------------------
__global__ void __launch_bounds__(32) attention(const float* __restrict__ Q,
                                                const float* __restrict__ K,
                                                const float* __restrict__ V,
                                                float* __restrict__ O) {
  __shared__ float pl[16 * 18];  // P tile, stride 18 keeps 8B alignment
  const int lane = threadIdx.x;
  const int m = lane & 15, hi = lane >> 4;
  const int qt = blockIdx.x;                 // query tile 0..63
  const int b = blockIdx.y / NHEADS;
  const int h = blockIdx.y % NHEADS;

  const float* qp = Q + (size_t)(b * TOK + qt * 16 + m) * DIMC + h * HDIM + 2 * hi;
  v2f aq[16];
#pragma unroll
  for (int c = 0; c < 16; c++) aq[c] = *(const v2f*)(qp + 4 * c);

  v8f oacc[4];
#pragma unroll
  for (int j = 0; j < 4; j++) oacc[j] = zero8();
  float mst[8], lst[8];
#pragma unroll
  for (int r = 0; r < 8; r++) { mst[r] = -1e30f; lst[r] = 0.0f; }

  for (int kt = 0; kt < TOK / 16; kt++) {
    // S tile = Q * K^T  (C layout: lane holds key-col m, rows r + 8*hi)
    const float* kp = K + (size_t)(b * TOK + kt * 16 + m) * DIMC + h * HDIM + 2 * hi;
    v8f s = zero8();
#pragma unroll
    for (int c = 0; c < 16; c++) {
      v2f bk = *(const v2f*)(kp + 4 * c);
      s = wmma4(aq[c], bk, s);
    }

    float alpha[8], prob[8];
#pragma unroll
    for (int r = 0; r < 8; r++) {
      float sv = s[r] * 0.125f;  // 1/sqrt(64)
      float mx = sv;
      mx = fmaxf(mx, __shfl_xor(mx, 1, 16));
      mx = fmaxf(mx, __shfl_xor(mx, 2, 16));
      mx = fmaxf(mx, __shfl_xor(mx, 4, 16));
      mx = fmaxf(mx, __shfl_xor(mx, 8, 16));
      float mnew = fmaxf(mst[r], mx);
      float al = expf(mst[r] - mnew);
      float pv = expf(sv - mnew);
      float sum = pv;
      sum += __shfl_xor(sum, 1, 16);
      sum += __shfl_xor(sum, 2, 16);
      sum += __shfl_xor(sum, 4, 16);
      sum += __shfl_xor(sum, 8, 16);
      mst[r] = mnew;
      lst[r] = lst[r] * al + sum;
      alpha[r] = al;
      prob[r] = pv;
    }
#pragma unroll
    for (int j = 0; j < 4; j++)
#pragma unroll
      for (int r = 0; r < 8; r++) oacc[j][r] *= alpha[r];

    // transpose P fragment via LDS: C layout -> A layout
    __syncthreads();
#pragma unroll
    for (int r = 0; r < 8; r++) pl[(r + 8 * hi) * 18 + m] = prob[r];
    __syncthreads();

    // O += P * V
    const float* vp = V + (size_t)(b * TOK + kt * 16) * DIMC + h * HDIM + m;
#pragma unroll
    for (int c = 0; c < 4; c++) {
      v2f pa = *(const v2f*)(&pl[m * 18 + 4 * c + 2 * hi]);
#pragma unroll
      for (int j = 0; j < 4; j++) {
        v2f bv;
        bv[0] = vp[(size_t)(4 * c + 2 * hi + 0) * DIMC + j * 16];
        bv[1] = vp[(size_t)(4 * c + 2 * hi + 1) * DIMC + j * 16];
        oacc[j] = wmma4(pa, bv, oacc[j]);
      }
    }
  }

  float* op = O + (size_t)(b * TOK + qt * 16) * DIMC + h * HDIM;
#pragma unroll
  for (int r = 0; r < 8; r++) {
    float inv = 1.0f / lst[r];
    int row = r + 8 * hi;
#pragma unroll
    for (int j = 0; j < 4; j++)
      op[(size_t)row * DIMC + j * 16 + m] = oacc[j][r] * inv;
  }
}

// ---------------------------------------------------------------------------
extern "C" void kernel_launch(void* const* d_in, const int* in_sizes, int n_in,
                              void* d_out, int out_size, void* d_ws, size_t ws_size,
                              hipStream_t stream) {
  const float* x  = (const float*)d_in[0];
  const float* vm = (const float*)d_in[1];
  const float* Ki = (const float*)d_in[2];
  const float* Wq = (const float*)d_in[3];
  const float* bq = (const float*)d_in[4];
  const float* Wk = (const float*)d_in[5];
  const float* bk = (const float*)d_in[6];
  const float* Wv = (const float*)d_in[7];
  const float* bv = (const float*)d_in[8];
  const float* Wo = (const float*)d_in[9];
  const float* bo = (const float*)d_in[10];
  float* out = (float*)d_out;

  float* ws = (float*)d_ws;
  float* P    = ws;                         // 512 floats
  float* Pinv = P + 512;                    // 512 floats
  float* Qb   = Pinv + 512;                 // BT*DIMC
  float* Kb   = Qb + (size_t)BT * DIMC;
  float* Vb   = Kb + (size_t)BT * DIMC;
  float* Ob   = Vb + (size_t)BT * DIMC;

  prep_mats<<<1, 32, 0, stream>>>(vm, Ki, P, Pinv);

  dim3 ggrid(BT / 16, DIMC / 64);
  gemm_xwt<<<ggrid, 32, 0, stream>>>(x, Wq, bq, Qb);
  gemm_xwt<<<ggrid, 32, 0, stream>>>(x, Wk, bk, Kb);
  gemm_xwt<<<ggrid, 32, 0, stream>>>(x, Wv, bv, Vb);

  int nth = BATCH * TOK * NHEADS;
  int tb = 256, tg = (nth + tb - 1) / tb;
  transform_tokens<<<tg, tb, 0, stream>>>(Qb, Pinv, 1, 1.0f);   // q: Pinv^T, +sin
  transform_tokens<<<tg, tb, 0, stream>>>(Kb, P, 0, 1.0f);      // k: P, +sin
  transform_tokens<<<tg, tb, 0, stream>>>(Vb, P, 0, 1.0f);      // v: P, +sin

  dim3 agrid(TOK / 16, BATCH * NHEADS);
  attention<<<agrid, 32, 0, stream>>>(Qb, Kb, Vb, Ob);

  transform_tokens<<<tg, tb, 0, stream>>>(Ob, Pinv, 0, -1.0f);  // o: Pinv, -sin

  gemm_xwt<<<ggrid, 32, 0, stream>>>(Ob, Wo, bo, out);
}